// SpanConsistencyLayer_19026705121552
// MI455X (gfx1250) — compile-verified
//
#include <hip/hip_runtime.h>
#include <stdint.h>
#include <math.h>

#define N_COLS        8192
#define TILE          2048
#define TILES_PER_ROW (N_COLS / TILE)     // 4
#define HALO          14                  // MAX_SPAN_WIDTH - 1
#define PADDED        (TILE + 2 * HALO)   // 2076
#define THREADS       256
#define ELEMS         (TILE / THREADS)    // 8
#define WREG          (ELEMS + 2 * HALO)  // 36
#define MAXW          15
#define NEG_INF_V     (-10000.0f)

// gfx1250 async global->LDS copy (ASYNCcnt-tracked; CDNA5 ISA).
// VDST operand is a VGPR holding the LDS byte address == low 32 bits of the flat
// shared-aperture address (aperture base low bits are zero).
__device__ __forceinline__ void async_copy_f32(const float* g, float* lds_ptr) {
  uint32_t loff  = (uint32_t)(uintptr_t)lds_ptr;
  uint64_t gaddr = (uint64_t)(uintptr_t)g;
  asm volatile("global_load_async_to_lds_b32 %0, %1, off"
               :: "v"(loff), "v"(gaddr)
               : "memory");
}

__device__ __forceinline__ void wait_async_lds() {
  asm volatile("s_wait_asynccnt 0" ::: "memory");
}

__global__ __launch_bounds__(THREADS)
void span_boost_kernel(const float* __restrict__ scores,
                       const int*  __restrict__ mask,
                       const float* __restrict__ wlogits,
                       const float* __restrict__ gamma,
                       float* __restrict__ out) {
  __shared__ __align__(16) float sm[PADDED];

  const int t          = threadIdx.x;
  const int row        = blockIdx.x / TILES_PER_ROW;
  const int tile       = blockIdx.x % TILES_PER_ROW;
  const int tile_start = tile * TILE;
  const size_t row_off = (size_t)row * N_COLS;

  // ---- Phase 1: async DMA the score tile (with halo) into LDS. Addresses are
  // clamped into the row (halo values are overwritten with -inf in phase 2).
  #pragma unroll
  for (int k = 0; k < (PADDED + THREADS - 1) / THREADS; ++k) {
    int j = t + k * THREADS;
    if (j < PADDED) {
      int col = tile_start - HALO + j;
      int cc  = col < 0 ? 0 : (col > N_COLS - 1 ? N_COLS - 1 : col);
      async_copy_f32(scores + row_off + cc, &sm[j]);
    }
  }

  // ---- Overlapped with the DMA: softmax over the 15 width logits.
  // Widths 2..15 use weight indices 0..13; index 14 only affects the denominator.
  float wl[MAXW];
  #pragma unroll
  for (int i = 0; i < MAXW; ++i) wl[i] = wlogits[i];
  float mx = wl[0];
  #pragma unroll
  for (int i = 1; i < MAXW; ++i) mx = fmaxf(mx, wl[i]);
  float ex[MAXW];
  float sum = 0.0f;
  #pragma unroll
  for (int i = 0; i < MAXW; ++i) { ex[i] = expf(wl[i] - mx); sum += ex[i]; }
  float ws[MAXW];
  #pragma unroll
  for (int i = 0; i < MAXW; ++i) ws[i] = ex[i] / sum;

  const float gam = gamma[0];

  wait_async_lds();

  // ---- Phase 2: apply mask in place; out-of-row halo -> -inf so spans that
  // fall outside [0,N) can never win any max. Each thread touches exactly the
  // slots it async-loaded, so no barrier is needed before this pass.
  #pragma unroll
  for (int k = 0; k < (PADDED + THREADS - 1) / THREADS; ++k) {
    int j = t + k * THREADS;
    if (j < PADDED) {
      int col = tile_start - HALO + j;
      float v;
      if (col < 0 || col >= N_COLS) {
        v = -INFINITY;
      } else {
        v = (mask[row_off + col] == 0) ? NEG_INF_V : sm[j];
      }
      sm[j] = v;
    }
  }
  __syncthreads();

  // ---- Phase 3: register-resident sliding window per thread.
  // Thread t owns output cols [tile_start + t*8, +8); its windows need
  // sm[t*8 .. t*8+35] -> nine aligned ds_load_b128.
  float W[WREG];
  const float4* s4 = (const float4*)&sm[t * ELEMS];
  #pragma unroll
  for (int q = 0; q < WREG / 4; ++q) {
    float4 x = s4[q];
    W[4 * q + 0] = x.x; W[4 * q + 1] = x.y;
    W[4 * q + 2] = x.z; W[4 * q + 3] = x.w;
  }

  const int col0 = tile_start + t * ELEMS;
  float res[ELEMS];
  #pragma unroll
  for (int e = 0; e < ELEMS; ++e) {
    // Prefix mins left/right of center (both include the center element).
    float L[HALO + 1], R[HALO + 1];
    L[0] = W[e + HALO];
    R[0] = L[0];
    #pragma unroll
    for (int d = 1; d <= HALO; ++d) {
      L[d] = fminf(L[d - 1], W[e + HALO - d]);
      R[d] = fminf(R[d - 1], W[e + HALO + d]);
    }
    // boost = max(masked, max over spans [i-p, i+q], 1 <= p+q <= 14,
    //             of ww[p+q-1] * min(L_p, R_q))
    float boost = W[e + HALO];
    #pragma unroll
    for (int s = 1; s <= HALO; ++s) {
      float gbest = fminf(L[0], R[s]);
      #pragma unroll
      for (int p = 1; p <= s; ++p)
        gbest = fmaxf(gbest, fminf(L[p], R[s - p]));
      boost = fmaxf(boost, __fmul_rn(gbest, ws[s - 1]));
    }
    res[e] = __fadd_rn(scores[row_off + col0 + e], __fmul_rn(gam, boost));
  }

  float4* o4 = (float4*)(out + row_off + col0);
  o4[0] = make_float4(res[0], res[1], res[2], res[3]);
  o4[1] = make_float4(res[4], res[5], res[6], res[7]);
}

extern "C" void kernel_launch(void* const* d_in, const int* in_sizes, int n_in,
                              void* d_out, int out_size, void* d_ws, size_t ws_size,
                              hipStream_t stream) {
  (void)n_in; (void)out_size; (void)d_ws; (void)ws_size;
  const float* scores = (const float*)d_in[0];
  const int*   mask   = (const int*)d_in[1];
  const float* wlog   = (const float*)d_in[2];
  const float* gamma  = (const float*)d_in[3];
  float*       outp   = (float*)d_out;

  const int B = in_sizes[0] / N_COLS;     // 256 for the reference shapes
  dim3 grid(B * TILES_PER_ROW);           // 1024 blocks, 8 wave32 waves each
  span_boost_kernel<<<grid, THREADS, 0, stream>>>(scores, mask, wlog, gamma, outp);
}